// GCN_13305808683527
// MI455X (gfx1250) — compile-verified
//
#include <hip/hip_runtime.h>

// ---------------------------------------------------------------------------
// GCN forward for MI455X (gfx1250, wave32).
// Aggregations are L2-atomic bound; GEMMs use V_WMMA_F32_16X16X4_F32 (fp32
// preserved -- matrix math is ~3.4 GFLOP, negligible vs scatter traffic).
// A-tiles staged to LDS via GLOBAL_LOAD_ASYNC_TO_LDS_B128 when available.
// ---------------------------------------------------------------------------

typedef float v2f __attribute__((ext_vector_type(2)));
typedef float v8f __attribute__((ext_vector_type(8)));
typedef int   v4i __attribute__((ext_vector_type(4)));

#define IN_F   64
#define HID1   128
#define HID2   64
#define NCLS   10

#if defined(__gfx1250__) && \
    __has_builtin(__builtin_amdgcn_global_load_async_to_lds_b128) && \
    __has_builtin(__builtin_amdgcn_s_wait_asynccnt)
#define USE_ASYNC_LDS 1
#else
#define USE_ASYNC_LDS 0
#endif

#if USE_ASYNC_LDS
#define GAS1 __attribute__((address_space(1)))
#define LAS3 __attribute__((address_space(3)))
static __device__ __forceinline__ void async_copy16(const float* g, float* l) {
    // signature: (v4i AS1* src, v4i AS3* dst, imm offset, imm cpol)
    __builtin_amdgcn_global_load_async_to_lds_b128(
        (GAS1 v4i*)g, (LAS3 v4i*)l, 0, 0);
}
static __device__ __forceinline__ void async_wait0() {
    __builtin_amdgcn_s_wait_asynccnt(0);
}
#else
static __device__ __forceinline__ void async_copy16(const float* g, float* l) {
    *(float4*)l = *(const float4*)g;
}
static __device__ __forceinline__ void async_wait0() {}
#endif

static __device__ __forceinline__ v8f wmma_f32(v2f a, v2f b, v8f c) {
    // 8 args: (neg_a, A, neg_b, B, c_mod, C, reuse_a, reuse_b)
    return __builtin_amdgcn_wmma_f32_16x16x4_f32(
        false, a, false, b, (short)0, c, false, false);
}

// ---------------------------------------------------------------------------
// Edge scatter: half-wave (16 lanes x float4 = 64 floats) per edge.
// Optionally accumulates degree (first pass only).
// ---------------------------------------------------------------------------
__global__ void agg64_kernel(const float* __restrict__ src,
                             const int* __restrict__ row,
                             const int* __restrict__ col,
                             float* __restrict__ dst,
                             float* __restrict__ deg,
                             int n_edges) {
    const int g   = threadIdx.x >> 4;      // edge slot in block (0..15)
    const int sub = threadIdx.x & 15;      // feature chunk (0..15)
    const int e   = blockIdx.x * 16 + g;
    if (e >= n_edges) return;
    const int r = row[e];
    const int c = col[e];
    const float4 v = *(const float4*)(src + (size_t)r * IN_F + sub * 4);
    float* o = dst + (size_t)c * IN_F + sub * 4;
    atomicAdd(o + 0, v.x);
    atomicAdd(o + 1, v.y);
    atomicAdd(o + 2, v.z);
    atomicAdd(o + 3, v.w);
    if (deg != nullptr && sub == 0) atomicAdd(deg + c, 1.0f);
}

// deg -> 1/max(deg,1) in place
__global__ void invdeg_kernel(float* __restrict__ deg, int n) {
    const int i = blockIdx.x * blockDim.x + threadIdx.x;
    if (i < n) deg[i] = 1.0f / fmaxf(deg[i], 1.0f);
}

// ---------------------------------------------------------------------------
// GEMM1: h1[M,128] = relu( (agg1[M,64] @ W1[64,128]) * inv_deg[row] + b1 )
// Block = 256 threads = 8 waves; block owns 16 rows, wave w owns N-tile w.
// A tile (16x64 f32 = 4KB) + 16 inv_deg values staged in LDS.
// ---------------------------------------------------------------------------
__global__ void gemm1_kernel(const float* __restrict__ A,
                             const float* __restrict__ B,
                             const float* __restrict__ bias,
                             const float* __restrict__ invdeg,
                             float* __restrict__ C, int M) {
    __shared__ float As[16 * IN_F];
    __shared__ float Sinv[16];
    const int tileM = blockIdx.x;          // 16-row stripe
    {
        const int idx = threadIdx.x * 4;   // 256 thr * float4 = 1024 f32
        int r = tileM * 16 + (idx >> 6);
        if (r > M - 1) r = M - 1;
        async_copy16(A + (size_t)r * IN_F + (idx & 63), As + idx);
        if (threadIdx.x < 16) {
            int rr = tileM * 16 + threadIdx.x;
            if (rr > M - 1) rr = M - 1;
            Sinv[threadIdx.x] = invdeg[rr];
        }
    }
    async_wait0();
    __syncthreads();

    const int wave = threadIdx.x >> 5;     // N-tile 0..7
    const int lane = threadIdx.x & 31;
    const int hl = lane & 15, hi = lane >> 4;

    v8f acc = {0.f, 0.f, 0.f, 0.f, 0.f, 0.f, 0.f, 0.f};
#pragma unroll
    for (int kk = 0; kk < IN_F / 4; ++kk) {
        const int ka = kk * 4 + 2 * hi;    // lanes 0-15: K0/K1, 16-31: K2/K3
        v2f a;
        a.x = As[hl * IN_F + ka];
        a.y = As[hl * IN_F + ka + 1];
        v2f b;
        b.x = B[(size_t)ka * HID1 + wave * 16 + hl];
        b.y = B[(size_t)(ka + 1) * HID1 + wave * 16 + hl];
        acc = wmma_f32(a, b, acc);
    }

    const int colN = wave * 16 + hl;
    const float bb = bias[colN];
    float* Cr = C + (size_t)tileM * 16 * HID1 + colN;
    if (tileM * 16 + 16 <= M) {            // wave-uniform: no tail
#pragma unroll
        for (int v = 0; v < 8; ++v) {
            const int rr = v + 8 * hi;
            Cr[(size_t)rr * HID1] = fmaxf(acc[v] * Sinv[rr] + bb, 0.0f);
        }
    } else {
#pragma unroll
        for (int v = 0; v < 8; ++v) {
            const int rr = v + 8 * hi;
            if (tileM * 16 + rr < M)
                Cr[(size_t)rr * HID1] = fmaxf(acc[v] * Sinv[rr] + bb, 0.0f);
        }
    }
}

// ---------------------------------------------------------------------------
// GEMM2: t[M,64] = h1[M,128] @ W2[128,64]   (bias/relu deferred past agg2)
// Block owns 32 rows (2 M-tiles x 4 N-tiles across 8 waves); A in LDS (16KB).
// ---------------------------------------------------------------------------
__global__ void gemm2_kernel(const float* __restrict__ A,
                             const float* __restrict__ B,
                             float* __restrict__ C, int M) {
    __shared__ float As[32 * HID1];
    const int blkM = blockIdx.x * 32;
    {
#pragma unroll
        for (int i = 0; i < 4; ++i) {      // 256 thr * 4 * float4 = 4096 f32
            const int idx = (threadIdx.x + i * 256) * 4;
            int r = blkM + (idx >> 7);
            if (r > M - 1) r = M - 1;
            async_copy16(A + (size_t)r * HID1 + (idx & 127), As + idx);
        }
    }
    async_wait0();
    __syncthreads();

    const int wave = threadIdx.x >> 5;
    const int tM = wave >> 2;              // 0..1
    const int tN = wave & 3;               // 0..3
    const int lane = threadIdx.x & 31;
    const int hl = lane & 15, hi = lane >> 4;

    v8f acc = {0.f, 0.f, 0.f, 0.f, 0.f, 0.f, 0.f, 0.f};
#pragma unroll
    for (int kk = 0; kk < HID1 / 4; ++kk) {
        const int ka = kk * 4 + 2 * hi;
        v2f a;
        a.x = As[(tM * 16 + hl) * HID1 + ka];
        a.y = As[(tM * 16 + hl) * HID1 + ka + 1];
        v2f b;
        b.x = B[(size_t)ka * HID2 + tN * 16 + hl];
        b.y = B[(size_t)(ka + 1) * HID2 + tN * 16 + hl];
        acc = wmma_f32(a, b, acc);
    }

    const int colN = tN * 16 + hl;
    const int rbase = blkM + tM * 16;
    float* Cr = C + (size_t)rbase * HID2 + colN;
    if (rbase + 16 <= M) {                 // wave-uniform: no tail
#pragma unroll
        for (int v = 0; v < 8; ++v) {
            const int rr = v + 8 * hi;
            Cr[(size_t)rr * HID2] = acc[v];
        }
    } else {
#pragma unroll
        for (int v = 0; v < 8; ++v) {
            const int rr = v + 8 * hi;
            if (rbase + rr < M) Cr[(size_t)rr * HID2] = acc[v];
        }
    }
}

// ---------------------------------------------------------------------------
// GEMM3: out[M,10] = relu(agg2*inv_deg + b2) @ W3[64,10] + b3
// Layer-2 epilogue fused into the A load; N padded to 16 (B cols >=10 zero).
// One wave per 16-row tile; whole idle waves exit before any WMMA/barrier.
// ---------------------------------------------------------------------------
__global__ void gemm3_kernel(const float* __restrict__ A,
                             const float* __restrict__ invdeg,
                             const float* __restrict__ b2,
                             const float* __restrict__ W3,
                             const float* __restrict__ b3,
                             float* __restrict__ out, int M) {
    const int wave  = threadIdx.x >> 5;
    const int tileM = blockIdx.x * 8 + wave;
    if (tileM * 16 >= M) return;           // wave-uniform exit

    const int lane = threadIdx.x & 31;
    const int hl = lane & 15, hi = lane >> 4;
    const int rowA = tileM * 16 + hl;
    const float s = invdeg[rowA];
    const bool live = (hl < NCLS);

    v8f acc = {0.f, 0.f, 0.f, 0.f, 0.f, 0.f, 0.f, 0.f};
#pragma unroll
    for (int kk = 0; kk < HID2 / 4; ++kk) {
        const int ka = kk * 4 + 2 * hi;
        v2f a;   // fused layer-2 epilogue: relu(agg2*inv_deg + b2)
        a.x = fmaxf(A[(size_t)rowA * HID2 + ka] * s + b2[ka], 0.0f);
        a.y = fmaxf(A[(size_t)rowA * HID2 + ka + 1] * s + b2[ka + 1], 0.0f);
        v2f b;
        b.x = live ? W3[(size_t)ka * NCLS + hl] : 0.0f;
        b.y = live ? W3[(size_t)(ka + 1) * NCLS + hl] : 0.0f;
        acc = wmma_f32(a, b, acc);         // EXEC full here; divergence is
    }                                      // only inside the selects above

    if (live) {
        const float bb = b3[hl];
#pragma unroll
        for (int v = 0; v < 8; ++v) {
            const int rowM = tileM * 16 + v + 8 * hi;
            out[(size_t)rowM * NCLS + hl] = acc[v] + bb;
        }
    }
}

// ---------------------------------------------------------------------------
extern "C" void kernel_launch(void* const* d_in, const int* in_sizes, int n_in,
                              void* d_out, int out_size, void* d_ws, size_t ws_size,
                              hipStream_t stream) {
    const float* x  = (const float*)d_in[0];
    const int*   ei = (const int*)d_in[1];
    // d_in[2] = n_nodes scalar (unused; derived from sizes)
    const float* W1 = (const float*)d_in[3];
    const float* b1 = (const float*)d_in[4];
    const float* W2 = (const float*)d_in[5];
    const float* b2 = (const float*)d_in[6];
    const float* W3 = (const float*)d_in[7];
    const float* b3 = (const float*)d_in[8];
    float* out = (float*)d_out;

    const int n_nodes = in_sizes[0] / IN_F;
    const int n_edges = in_sizes[1] / 2;
    const int* row = ei;
    const int* col = ei + n_edges;

    // Workspace layout (floats): deg | agg1 | h1 | t2 | agg2
    float* ws = (float*)d_ws;
    size_t off = 0;
    float* deg  = ws + off; off += (size_t)n_nodes;
    float* agg1 = ws + off; off += (size_t)n_nodes * IN_F;
    float* h1   = ws + off; off += (size_t)n_nodes * HID1;
    float* t2   = ws + off; off += (size_t)n_nodes * HID2;
    float* agg2 = ws + off; off += (size_t)n_nodes * HID2;

    // Accumulated buffers must be zeroed every call (graph-capture legal).
    (void)hipMemsetAsync(deg,  0, sizeof(float) * (size_t)n_nodes, stream);
    (void)hipMemsetAsync(agg1, 0, sizeof(float) * (size_t)n_nodes * IN_F, stream);
    (void)hipMemsetAsync(agg2, 0, sizeof(float) * (size_t)n_nodes * HID2, stream);

    const int eblocks = (n_edges + 15) / 16;

    // Layer 1: aggregate x (64-wide) + degree, then fused GEMM epilogue.
    agg64_kernel<<<eblocks, 256, 0, stream>>>(x, row, col, agg1, deg, n_edges);
    invdeg_kernel<<<(n_nodes + 255) / 256, 256, 0, stream>>>(deg, n_nodes);
    gemm1_kernel<<<(n_nodes + 15) / 16, 256, 0, stream>>>(agg1, W1, b1, deg, h1, n_nodes);

    // Layer 2 reordered: GEMM (128->64) BEFORE aggregation -> halves traffic.
    gemm2_kernel<<<(n_nodes + 31) / 32, 256, 0, stream>>>(h1, W2, t2, n_nodes);
    agg64_kernel<<<eblocks, 256, 0, stream>>>(t2, row, col, agg2, nullptr, n_edges);

    // Layer 3 with fused layer-2 epilogue (inv_deg, +b2, relu) on A-load.
    gemm3_kernel<<<((n_nodes + 15) / 16 + 7) / 8, 256, 0, stream>>>(
        agg2, deg, b2, W3, b3, out, n_nodes);
}